// Decoder_65738769433295
// MI455X (gfx1250) — compile-verified
//
#include <hip/hip_runtime.h>
#include <math.h>

#define VSZ 50257
#define HSZ 1024

typedef __attribute__((ext_vector_type(2))) float v2f;
typedef __attribute__((ext_vector_type(8))) float v8f;

__device__ __forceinline__ float sigm(float x) { return 1.0f / (1.0f + expf(-x)); }

// ---------------------------------------------------------------------------
// K1: gates[r] = e . w_ih[r] + h . w_hh[r] + b_ih[r] + b_hh[r],  r in [0,4H)
// One wave computes 16 rows via V_WMMA_F32_16X16X4_F32, A = broadcast vector.
// ---------------------------------------------------------------------------
__global__ __launch_bounds__(256) void lstm_gates_wmma(
    const int* __restrict__ x,
    const float* __restrict__ emb,
    const float* __restrict__ w_ih,
    const float* __restrict__ w_hh,
    const float* __restrict__ b_ih,
    const float* __restrict__ b_hh,
    const float* __restrict__ h_old,
    float* __restrict__ gates)
{
    const int lane  = threadIdx.x & 31;
    const int wave  = threadIdx.x >> 5;
    const int gwave = blockIdx.x * 8 + wave;
    const int row_base = gwave * 16;              // 16 gate rows per wave
    if (row_base >= 4 * HSZ) return;              // uniform per wave

    const int n  = lane & 15;                     // output column N
    const int kh = lane >> 4;                     // half: K offset 0 or 2

    const float* erow = emb + (size_t)x[0] * HSZ;
    const float* wih  = w_ih + (size_t)(row_base + n) * HSZ + 2 * kh;
    const float* whh  = w_hh + (size_t)(row_base + n) * HSZ + 2 * kh;

    v8f acc = {};
#pragma unroll 4
    for (int k0 = 0; k0 < HSZ; k0 += 4) {
        v2f ae, ah;
        ae[0] = erow[k0 + 2 * kh];  ae[1] = erow[k0 + 2 * kh + 1];
        ah[0] = h_old[k0 + 2 * kh]; ah[1] = h_old[k0 + 2 * kh + 1];
        v2f bi = *(const v2f*)(wih + k0);
        v2f bh = *(const v2f*)(whh + k0);
        acc = __builtin_amdgcn_wmma_f32_16x16x4_f32(false, ae, false, bi,
                                                    (short)0, acc, false, false);
        acc = __builtin_amdgcn_wmma_f32_16x16x4_f32(false, ah, false, bh,
                                                    (short)0, acc, false, false);
    }
    if (lane < 16) {
        const int r = row_base + lane;            // D[0][N=lane] is acc[0] of lane
        gates[r] = acc[0] + b_ih[r] + b_hh[r];
    }
}

// ---------------------------------------------------------------------------
// K2: pointwise LSTM cell update (PyTorch gate order i,f,g,o)
// ---------------------------------------------------------------------------
__global__ __launch_bounds__(256) void lstm_pointwise(
    const float* __restrict__ gates,
    const float* __restrict__ c_old,
    float* __restrict__ h_ws,
    float* __restrict__ h_out,
    float* __restrict__ c_out)
{
    const int j = blockIdx.x * blockDim.x + threadIdx.x;
    if (j < HSZ) {
        const float ig = sigm(gates[j]);
        const float fg = sigm(gates[HSZ + j]);
        const float gg = tanhf(gates[2 * HSZ + j]);
        const float og = sigm(gates[3 * HSZ + j]);
        const float c  = fg * c_old[j] + ig * gg;
        const float h  = og * tanhf(c);
        h_ws[j]  = h;
        h_out[j] = h;
        c_out[j] = c;
    }
}

// ---------------------------------------------------------------------------
// K3: logits[r] = h_new . fc_w[r] + fc_b[r]   (streams 206MB of fc_w once)
// 16 vocab rows per wave via WMMA; also emits per-block max for log-softmax.
// ---------------------------------------------------------------------------
__global__ __launch_bounds__(256) void fc_logits_wmma(
    const float* __restrict__ fc_w,
    const float* __restrict__ fc_b,
    const float* __restrict__ h_new,
    float* __restrict__ logits,       // == d_out[0..V)
    float* __restrict__ blockmax)
{
    __shared__ float smax[8];
    const int lane  = threadIdx.x & 31;
    const int wave  = threadIdx.x >> 5;
    const int gwave = blockIdx.x * 8 + wave;
    const int row_base = gwave * 16;

    float myval = -INFINITY;
    if (row_base < VSZ) {                         // uniform per wave
        const int n  = lane & 15;
        const int kh = lane >> 4;
        int rown = row_base + n;
        if (rown >= VSZ) rown = VSZ - 1;          // clamp loads; store is guarded
        const float* wrow = fc_w + (size_t)rown * HSZ + 2 * kh;

        v8f acc = {};
#pragma unroll 4
        for (int k0 = 0; k0 < HSZ; k0 += 4) {
            v2f a;
            a[0] = h_new[k0 + 2 * kh];
            a[1] = h_new[k0 + 2 * kh + 1];
            v2f b = *(const v2f*)(wrow + k0);
            acc = __builtin_amdgcn_wmma_f32_16x16x4_f32(false, a, false, b,
                                                        (short)0, acc, false, false);
        }
        const int r = row_base + lane;
        if (lane < 16 && r < VSZ) {
            const float v = acc[0] + fc_b[r];
            logits[r] = v;
            myval = v;
        }
    }
    // wave-level max, then block-level max
    for (int off = 16; off; off >>= 1)
        myval = fmaxf(myval, __shfl_down(myval, off, 32));
    if (lane == 0) smax[wave] = myval;
    __syncthreads();
    if (threadIdx.x == 0) {
        float m = smax[0];
        for (int i = 1; i < 8; ++i) m = fmaxf(m, smax[i]);
        blockmax[blockIdx.x] = m;
    }
}

// ---------------------------------------------------------------------------
// K4: single block: M = max(blockmax), S = sum exp(logit - M), L = M + log S
// ---------------------------------------------------------------------------
__global__ __launch_bounds__(512) void softmax_stats(
    const float* __restrict__ logits,
    const float* __restrict__ blockmax,
    int nblocks,
    float* __restrict__ scalarL)
{
    __shared__ float sred[16];
    const int tid  = threadIdx.x;
    const int lane = tid & 31;
    const int wv   = tid >> 5;                    // 16 waves

    float m = -INFINITY;
    for (int i = tid; i < nblocks; i += blockDim.x) m = fmaxf(m, blockmax[i]);
    for (int off = 16; off; off >>= 1) m = fmaxf(m, __shfl_down(m, off, 32));
    if (lane == 0) sred[wv] = m;
    __syncthreads();
    if (wv == 0) {
        float t = (lane < 16) ? sred[lane] : -INFINITY;
        for (int off = 16; off; off >>= 1) t = fmaxf(t, __shfl_down(t, off, 32));
        if (lane == 0) sred[0] = t;
    }
    __syncthreads();
    const float M = sred[0];

    float s = 0.0f;
    for (int i = tid; i < VSZ; i += blockDim.x) s += expf(logits[i] - M);
    __syncthreads();                              // done reading sred[0]
    for (int off = 16; off; off >>= 1) s += __shfl_down(s, off, 32);
    if (lane == 0) sred[wv] = s;
    __syncthreads();
    if (tid == 0) {
        float tot = 0.0f;
        for (int i = 0; i < 16; ++i) tot += sred[i];
        scalarL[0] = M + logf(tot);
    }
}

// ---------------------------------------------------------------------------
// K5: in-place log-softmax: d_out[v] -= L
// ---------------------------------------------------------------------------
__global__ __launch_bounds__(256) void logsoftmax_apply(
    float* __restrict__ out, const float* __restrict__ scalarL)
{
    const int v = blockIdx.x * blockDim.x + threadIdx.x;
    if (v < VSZ) out[v] -= scalarL[0];
}

// ---------------------------------------------------------------------------
extern "C" void kernel_launch(void* const* d_in, const int* in_sizes, int n_in,
                              void* d_out, int out_size, void* d_ws, size_t ws_size,
                              hipStream_t stream)
{
    const int*   x      = (const int*)d_in[0];
    const float* hidden = (const float*)d_in[1];
    const float* cell   = (const float*)d_in[2];
    const float* emb    = (const float*)d_in[3];
    const float* w_ih   = (const float*)d_in[4];
    const float* w_hh   = (const float*)d_in[5];
    const float* b_ih   = (const float*)d_in[6];
    const float* b_hh   = (const float*)d_in[7];
    const float* fc_w   = (const float*)d_in[8];
    const float* fc_b   = (const float*)d_in[9];
    float* out = (float*)d_out;

    float* ws       = (float*)d_ws;
    float* ws_gates = ws;            // 4096 floats
    float* ws_h     = ws + 4096;     // 1024 floats
    float* ws_bmax  = ws + 5120;     // up to 512 floats
    float* ws_L     = ws + 5632;     // 1 float

    // K1: 4096 gate rows, 16 rows/wave, 8 waves/block -> 32 blocks
    lstm_gates_wmma<<<32, 256, 0, stream>>>(x, emb, w_ih, w_hh, b_ih, b_hh,
                                            hidden, ws_gates);
    // K2: pointwise cell update; writes h_new/c_new into d_out tail + ws
    lstm_pointwise<<<(HSZ + 255) / 256, 256, 0, stream>>>(
        ws_gates, cell, ws_h, out + VSZ, out + VSZ + HSZ);
    // K3: vocab projection, 16 rows/wave, 8 waves/block
    const int FC_BLOCKS = (VSZ + 16 * 8 - 1) / (16 * 8);   // 393
    fc_logits_wmma<<<FC_BLOCKS, 256, 0, stream>>>(fc_w, fc_b, ws_h, out, ws_bmax);
    // K4: softmax statistics (single block)
    softmax_stats<<<1, 512, 0, stream>>>(out, ws_bmax, FC_BLOCKS, ws_L);
    // K5: apply log-softmax shift in place
    logsoftmax_apply<<<(VSZ + 255) / 256, 256, 0, stream>>>(out, ws_L);
}